// NodeClassificationWgGNNModel_39986145526073
// MI455X (gfx1250) — compile-verified
//
#include <hip/hip_runtime.h>

// Problem constants (match reference)
#define IN_DIM 128
#define HID    256
#define NCLS   47
#define CN1    45056
#define CN2    4096
#define F0     25
#define F1     10

typedef float v2f __attribute__((ext_vector_type(2)));
typedef float v4f __attribute__((ext_vector_type(4)));
typedef float v8f __attribute__((ext_vector_type(8)));

// ---------------------------------------------------------------------------
// Gather + mean + concat for layer 0:
//   X0[r][0:128]   = mean_{j<25} node_feat[gid0[col0[r*25+j]]]
//   X0[r][128:256] = node_feat[gid0[r]]
// One wave32 per destination row; each lane owns one float4 chunk (128 f32).
// ---------------------------------------------------------------------------
__global__ void __launch_bounds__(256)
gather0_kernel(const float* __restrict__ nf, const int* __restrict__ gid0,
               const int* __restrict__ col0, float* __restrict__ X0) {
  const int r = blockIdx.x * 8 + (threadIdx.x >> 5);
  const int lane = threadIdx.x & 31;
  if (r >= CN1) return;
  const int* cp = col0 + (size_t)r * F0;
  v4f s = {0.f, 0.f, 0.f, 0.f};
#pragma unroll 5
  for (int j = 0; j < F0; ++j) {
    const int e = cp[j];
    const int g = gid0[e];
    s += *(const v4f*)(nf + (size_t)g * IN_DIM + lane * 4);
  }
  s *= (1.0f / (float)F0);
  float* xr = X0 + (size_t)r * (2 * IN_DIM);
  *(v4f*)(xr + lane * 4) = s;
  const int gs = gid0[r];
  *(v4f*)(xr + IN_DIM + lane * 4) =
      *(const v4f*)(nf + (size_t)gs * IN_DIM + lane * 4);
}

// ---------------------------------------------------------------------------
// Gather + mean + concat for layer 1:
//   X1[r][0:256]   = mean_{j<10} H[col1[r*10+j]]
//   X1[r][256:512] = H[r]
// One wave per row; each lane owns two float4 chunks (256 f32).
// ---------------------------------------------------------------------------
__global__ void __launch_bounds__(256)
gather1_kernel(const float* __restrict__ H, const int* __restrict__ col1,
               float* __restrict__ X1) {
  const int r = blockIdx.x * 8 + (threadIdx.x >> 5);
  const int lane = threadIdx.x & 31;
  if (r >= CN2) return;
  const int* cp = col1 + (size_t)r * F1;
  v4f s0 = {0.f, 0.f, 0.f, 0.f}, s1 = {0.f, 0.f, 0.f, 0.f};
#pragma unroll 5
  for (int j = 0; j < F1; ++j) {
    const float* h = H + (size_t)cp[j] * HID;
    s0 += *(const v4f*)(h + lane * 4);
    s1 += *(const v4f*)(h + 128 + lane * 4);
  }
  const float inv = 1.0f / (float)F1;
  s0 *= inv; s1 *= inv;
  float* xr = X1 + (size_t)r * (2 * HID);
  *(v4f*)(xr + lane * 4)       = s0;
  *(v4f*)(xr + 128 + lane * 4) = s1;
  const float* hs = H + (size_t)r * HID;
  *(v4f*)(xr + HID + lane * 4)       = *(const v4f*)(hs + lane * 4);
  *(v4f*)(xr + HID + 128 + lane * 4) = *(const v4f*)(hs + 128 + lane * 4);
}

// Stack Wn0 over Wr0 -> Wcat0 [256][256]
__global__ void pack_w0_kernel(const float* __restrict__ Wn,
                               const float* __restrict__ Wr,
                               float* __restrict__ Wc) {
  const int i = blockIdx.x * blockDim.x + threadIdx.x;
  if (i >= 256 * 256) return;
  const int k = i >> 8, n = i & 255;
  Wc[i] = (k < IN_DIM) ? Wn[(size_t)k * HID + n] : Wr[(size_t)(k - IN_DIM) * HID + n];
}

// Stack Wn1 over Wr1, pad 47 -> 48 cols -> Wcat1 [512][48]
__global__ void pack_w1_kernel(const float* __restrict__ Wn,
                               const float* __restrict__ Wr,
                               float* __restrict__ Wc) {
  const int i = blockIdx.x * blockDim.x + threadIdx.x;
  if (i >= 512 * 48) return;
  const int k = i / 48, n = i % 48;
  float v = 0.0f;
  if (n < NCLS)
    v = (k < HID) ? Wn[(size_t)k * NCLS + n] : Wr[(size_t)(k - HID) * NCLS + n];
  Wc[i] = v;
}

// ---------------------------------------------------------------------------
// f32 WMMA GEMM: C[M][ldc] = act(A[M][K] @ B[K][NB] + bias), V_WMMA_F32_16X16X4_F32.
// 8 waves/block; wave w computes 16x16 tile at rows blockIdx.x*128 + w*16,
// cols blockIdx.y*16. B panel (K x 16) staged in LDS, shared by all waves.
// A-fragment (lane view of 16x4 tile): lanes 0-15 hold K={k,k+1} of row M=lane,
// lanes 16-31 hold K={k+2,k+3}. B-fragment mirrors: lane half selects K pair,
// lane&15 selects N. Accumulator chained through SRC2 over K/4 steps.
// ---------------------------------------------------------------------------
template <int K, bool RELU>
__global__ void __launch_bounds__(256)
gemm_wmma_kernel(const float* __restrict__ A, const float* __restrict__ B, int NB,
                 const float* __restrict__ bias, float* __restrict__ C,
                 int ldc, int Nvalid) {
  __shared__ float sB[K * 16];
  const int tid  = threadIdx.x;
  const int col0 = blockIdx.y * 16;
  for (int idx = tid; idx < K * 16; idx += 256) {
    const int k = idx >> 4, n = idx & 15;
    sB[idx] = B[(size_t)k * NB + col0 + n];
  }
  __syncthreads();

  const int wave = tid >> 5;
  const int lane = tid & 31;
  const int half = lane >> 4;      // 0: K pair {0,1}; 1: K pair {2,3}
  const int mn   = lane & 15;      // A row within tile / B-C column within tile
  const int row0 = blockIdx.x * 128 + wave * 16;

  const float* Arow = A + (size_t)(row0 + mn) * K + 2 * half;
  v8f acc = {0.f, 0.f, 0.f, 0.f, 0.f, 0.f, 0.f, 0.f};

#pragma unroll 8
  for (int kk = 0; kk < K; kk += 4) {
    if ((kk & 63) == 0) __builtin_prefetch(Arow + kk + 128, 0, 0);
    const v2f a = *(const v2f*)(Arow + kk);
    v2f b;
    const int kb = (kk + 2 * half) * 16 + mn;
    b.x = sB[kb];
    b.y = sB[kb + 16];
    acc = __builtin_amdgcn_wmma_f32_16x16x4_f32(
        /*neg_a=*/false, a, /*neg_b=*/false, b,
        /*c_mod=*/(short)0, acc, /*reuse_a=*/false, /*reuse_b=*/false);
  }

  const int col = col0 + mn;
  const bool cok = (col < Nvalid);
  const float bv = cok ? bias[col] : 0.0f;
#pragma unroll
  for (int i = 0; i < 8; ++i) {
    float v = acc[i] + bv;
    if (RELU) v = fmaxf(v, 0.0f);
    if (cok) C[(size_t)(row0 + half * 8 + i) * ldc + col] = v;
  }
}

// ---------------------------------------------------------------------------
extern "C" void kernel_launch(void* const* d_in, const int* in_sizes, int n_in,
                              void* d_out, int out_size, void* d_ws, size_t ws_size,
                              hipStream_t stream) {
  const float* node_feat = (const float*)d_in[0];
  const int*   gid0      = (const int*)d_in[1];
  const int*   col0      = (const int*)d_in[2];
  const int*   col1      = (const int*)d_in[3];
  const float* Wn0       = (const float*)d_in[4];
  const float* Wr0       = (const float*)d_in[5];
  const float* b0        = (const float*)d_in[6];
  const float* Wn1       = (const float*)d_in[7];
  const float* Wr1       = (const float*)d_in[8];
  const float* b1        = (const float*)d_in[9];
  float* out = (float*)d_out;

  // Workspace carve-out (256B aligned slabs)
  char* ws = (char*)d_ws;
  size_t off = 0;
  auto carve = [&](size_t bytes) -> float* {
    float* p = (float*)(ws + off);
    off += (bytes + 255) & ~(size_t)255;
    return p;
  };
  float* X0  = carve((size_t)CN1 * 2 * IN_DIM * sizeof(float)); // 46.1 MB
  float* H   = carve((size_t)CN1 * HID * sizeof(float));        // 46.1 MB
  float* X1  = carve((size_t)CN2 * 2 * HID * sizeof(float));    //  8.4 MB
  float* W0c = carve((size_t)256 * 256 * sizeof(float));        //  256 KB
  float* W1c = carve((size_t)512 * 48 * sizeof(float));         //   98 KB
  (void)ws_size; (void)in_sizes; (void)n_in; (void)out_size;

  // Weight stacking (tiny)
  pack_w0_kernel<<<(256 * 256 + 255) / 256, 256, 0, stream>>>(Wn0, Wr0, W0c);
  pack_w1_kernel<<<(512 * 48 + 255) / 256, 256, 0, stream>>>(Wn1, Wr1, W1c);

  // Layer 0: gather/mean/concat, then [45056,256] @ [256,256] + b0, ReLU
  gather0_kernel<<<CN1 / 8, 256, 0, stream>>>(node_feat, gid0, col0, X0);
  gemm_wmma_kernel<256, true><<<dim3(CN1 / 128, HID / 16), 256, 0, stream>>>(
      X0, W0c, 256, b0, H, HID, HID);

  // Layer 1: gather/mean/concat, then [4096,512] @ [512,48] + b1 (store 47 cols)
  gather1_kernel<<<CN2 / 8, 256, 0, stream>>>(H, col1, X1);
  gemm_wmma_kernel<512, false><<<dim3(CN2 / 128, 48 / 16), 256, 0, stream>>>(
      X1, W1c, 48, b1, out, NCLS, NCLS);
}